// MGATConv_82102594830696
// MI455X (gfx1250) — compile-verified
//
#include <hip/hip_runtime.h>
#include <hip/hip_bf16.h>
#include <math.h>

// Problem constants (fixed by the reference)
#define DIN   128
#define H     8
#define F     32
#define HF    256      // H*F
#define TT    8        // edge types
#define NEG_SLOPE 0.2f

typedef __attribute__((ext_vector_type(2))) float v2f;
typedef __attribute__((ext_vector_type(8))) float v8f;

// ---------------------------------------------------------------------------
// Kernel 0: tiny precompute.
//   Me[h][j]  = sum_f attn_e[h,f] * W_e[h*F+f, j]          (8x8)
//   ts[t][h]  = sum_f attn_et[h,f] * dot(W_et[h*F+f,:], type_emb[t,:])  (8x8)
//   counts[]  = 0
// ---------------------------------------------------------------------------
__global__ __launch_bounds__(64) void precompute_kernel(
    const float* __restrict__ attn_e, const float* __restrict__ W_e,
    const float* __restrict__ attn_et, const float* __restrict__ W_et,
    const float* __restrict__ type_emb,
    float* __restrict__ Me, float* __restrict__ ts, int* __restrict__ counts) {
  int tid = threadIdx.x;           // 0..63
  if (tid < TT) counts[tid] = 0;
  int a = tid >> 3, b = tid & 7;   // (h,j) for Me, (t,h) for ts
  // Me
  float acc = 0.f;
  for (int f = 0; f < F; ++f)
    acc += attn_e[a * F + f] * W_e[(a * F + f) * H + b];
  Me[a * H + b] = acc;
  // ts  (a = type t, b = head h)
  float acc2 = 0.f;
  for (int f = 0; f < F; ++f) {
    float d = 0.f;
    for (int k = 0; k < F; ++k)
      d += W_et[(b * F + f) * F + k] * type_emb[a * F + k];
    acc2 += attn_et[b * F + f] * d;
  }
  ts[a * H + b] = acc2;
}

// ---------------------------------------------------------------------------
// Kernel 1: edge-type histogram (LDS local bins -> global atomics)
// ---------------------------------------------------------------------------
__global__ __launch_bounds__(256) void hist_kernel(
    const int* __restrict__ etype, int* __restrict__ counts, int E) {
  __shared__ int lh[TT];
  if (threadIdx.x < TT) lh[threadIdx.x] = 0;
  __syncthreads();
  int e = blockIdx.x * blockDim.x + threadIdx.x;
  if (e < E) atomicAdd(&lh[etype[e]], 1);
  __syncthreads();
  if (threadIdx.x < TT) atomicAdd(&counts[threadIdx.x], lh[threadIdx.x]);
}

// ---------------------------------------------------------------------------
// Kernel 2: fs = feat @ W_src^T  via V_WMMA_F32_16X16X4_F32 (fp32 precision).
//   feat: [N,128]   W: [256,128]   fs: [N,256]
//   grid = (N/16, 2), block = 256 (8 waves). Each wave -> one 16x16 D tile.
//   A 16x4 frag: lane l holds rows l&15, K = (l>>4)*2 + {0,1}
//   B 4x16 frag: lane l holds col  l&15, K = (l>>4)*2 + {0,1}  (B = W^T)
// ---------------------------------------------------------------------------
__global__ __launch_bounds__(256) void gemm_wmma_kernel(
    const float* __restrict__ feat, const float* __restrict__ W,
    float* __restrict__ fs) {
  int wave = threadIdx.x >> 5;
  int lane = threadIdx.x & 31;
  int row0 = blockIdx.x * 16;                 // node-row tile
  int col0 = (blockIdx.y * 8 + wave) * 16;    // output-col tile (0..255)
  int m    = lane & 15;
  int koff = (lane >> 4) << 1;                // 0 or 2

  const float* arow = feat + (size_t)(row0 + m) * DIN;
  const float* brow = W    + (size_t)(col0 + m) * DIN;

  v8f c = {};
  #pragma unroll 4
  for (int k = 0; k < DIN; k += 4) {
    v2f a = *(const v2f*)(arow + k + koff);
    v2f b = *(const v2f*)(brow + k + koff);
    c = __builtin_amdgcn_wmma_f32_16x16x4_f32(
            /*neg_a=*/false, a, /*neg_b=*/false, b,
            /*c_mod=*/(short)0, c, /*reuse_a=*/false, /*reuse_b=*/false);
  }

  int r_off = (lane >> 4) << 3;               // 0 or 8
  int cc    = lane & 15;
  #pragma unroll
  for (int i = 0; i < 8; ++i)
    fs[(size_t)(row0 + i + r_off) * HF + col0 + cc] = c[i];
}

// ---------------------------------------------------------------------------
// Kernel 3: el[n,h] = dot(fs[n,h,:], attn_l[h,:]);  er likewise.
// ---------------------------------------------------------------------------
__global__ __launch_bounds__(256) void el_er_kernel(
    const float* __restrict__ fs, const float* __restrict__ attn_l,
    const float* __restrict__ attn_r, float* __restrict__ el,
    float* __restrict__ er, int NH) {
  int idx = blockIdx.x * blockDim.x + threadIdx.x;   // n*H + h
  if (idx >= NH) return;
  int h = idx & (H - 1);
  const float* row = fs + (size_t)(idx >> 3) * HF + h * F;
  const float* al = attn_l + h * F;
  const float* ar = attn_r + h * F;
  float sl = 0.f, sr = 0.f;
  #pragma unroll
  for (int f = 0; f < F; ++f) {
    float v = row[f];
    sl += v * al[f];
    sr += v * ar[f];
  }
  el[idx] = sl;
  er[idx] = sr;
}

// ---------------------------------------------------------------------------
// Kernel 4: per-edge attention logits (pre-softmax), stored [E,H].
//   attn = leaky_relu( (el[src]+er[dst] + edge_emb.Me[h] + ts[etype,h]) * w[etype] )
//   w[t] = E / (T * counts[t])
// ---------------------------------------------------------------------------
__global__ __launch_bounds__(256) void edge_attn_kernel(
    const float* __restrict__ el, const float* __restrict__ er,
    const float* __restrict__ eemb, const int* __restrict__ src,
    const int* __restrict__ dst, const int* __restrict__ etype,
    const float* __restrict__ Me, const float* __restrict__ ts,
    const int* __restrict__ counts, float* __restrict__ attn, int E) {
  __shared__ float sMe[64], sTs[64], sW[TT];
  if (threadIdx.x < 64) {
    sMe[threadIdx.x] = Me[threadIdx.x];
    sTs[threadIdx.x] = ts[threadIdx.x];
  }
  if (threadIdx.x < TT)
    sW[threadIdx.x] = (float)E / ((float)TT * (float)counts[threadIdx.x]);
  __syncthreads();

  int e = blockIdx.x * blockDim.x + threadIdx.x;
  if (e >= E) return;
  int s = src[e], d = dst[e], t = etype[e];

  const float4* ep = (const float4*)(eemb + (size_t)e * H);
  float4 e0 = ep[0], e1 = ep[1];
  float emb[H] = {e0.x, e0.y, e0.z, e0.w, e1.x, e1.y, e1.z, e1.w};
  float w = sW[t];

  const float* els = el + (size_t)s * H;
  const float* erd = er + (size_t)d * H;
  float* out = attn + (size_t)e * H;
  #pragma unroll
  for (int h = 0; h < H; ++h) {
    float esc = 0.f;
    #pragma unroll
    for (int j = 0; j < H; ++j) esc += emb[j] * sMe[h * H + j];
    float a = (els[h] + erd[h] + esc + sTs[t * H + h]) * w;
    out[h] = (a >= 0.f) ? a : NEG_SLOPE * a;
  }
}

// ---------------------------------------------------------------------------
// Kernel 5: CSR row pointers from sorted dst (lower_bound binary search).
// ---------------------------------------------------------------------------
__global__ __launch_bounds__(256) void row_ptr_kernel(
    const int* __restrict__ dst, int* __restrict__ rowptr, int N, int E) {
  int n = blockIdx.x * blockDim.x + threadIdx.x;
  if (n > N) return;
  int lo = 0, hi = E;
  while (lo < hi) {
    int mid = (lo + hi) >> 1;
    if (dst[mid] < n) lo = mid + 1; else hi = mid;
  }
  rowptr[n] = lo;
}

// ---------------------------------------------------------------------------
// Kernel 6: per-node edge softmax + weighted aggregation.
//   block = 256 threads = 8 waves; wave w owns head h=w for the reductions;
//   in phase C thread tid owns output element (h,f) = (tid>>5, tid&31).
// ---------------------------------------------------------------------------
__global__ __launch_bounds__(256) void aggregate_kernel(
    const float* __restrict__ fs, const float* __restrict__ attn,
    const int* __restrict__ src, const int* __restrict__ rowptr,
    const float* __restrict__ bias, float* __restrict__ out) {
  int n    = blockIdx.x;
  int tid  = threadIdx.x;
  int h    = tid >> 5;
  int lane = tid & 31;
  int s = rowptr[n], t = rowptr[n + 1];
  float b = bias[tid];
  if (s == t) {                      // no incoming edges: out = bias
    out[(size_t)n * HF + tid] = b;
    return;
  }
  __shared__ float sm[H], sz[H];

  // Phase A: per-head max (wave h reduces over its segment)
  float m = -3.402823466e38f;
  for (int e = s + lane; e < t; e += 32)
    m = fmaxf(m, attn[(size_t)e * H + h]);
  #pragma unroll
  for (int off = 16; off; off >>= 1) m = fmaxf(m, __shfl_xor(m, off, 32));
  if (lane == 0) sm[h] = m;
  __syncthreads();
  m = sm[h];

  // Phase B: per-head sum of exp
  float z = 0.f;
  for (int e = s + lane; e < t; e += 32)
    z += __expf(attn[(size_t)e * H + h] - m);
  #pragma unroll
  for (int off = 16; off; off >>= 1) z += __shfl_xor(z, off, 32);
  if (lane == 0) sz[h] = z;
  __syncthreads();
  float inv_z = 1.0f / sz[h];

  // Phase C: out[n,h,f] = bias + sum_e softmax(e,h) * fs[src[e], h, f]
  float acc = 0.f;
  for (int e = s; e < t; ++e) {
    float coeff = __expf(attn[(size_t)e * H + h] - m) * inv_z;
    int sr = src[e];
    if (e + 1 < t)
      __builtin_prefetch(fs + (size_t)src[e + 1] * HF + tid, 0, 0);
    acc += coeff * fs[(size_t)sr * HF + tid];
  }
  out[(size_t)n * HF + tid] = acc + b;
}

// ---------------------------------------------------------------------------
// Host-side orchestration
// ---------------------------------------------------------------------------
extern "C" void kernel_launch(void* const* d_in, const int* in_sizes, int n_in,
                              void* d_out, int out_size, void* d_ws, size_t ws_size,
                              hipStream_t stream) {
  const float* feat     = (const float*)d_in[0];
  const float* eemb     = (const float*)d_in[1];
  const int*   src      = (const int*)d_in[2];
  const int*   dst      = (const int*)d_in[3];
  const int*   etype    = (const int*)d_in[4];
  const float* W_src    = (const float*)d_in[5];
  const float* attn_l   = (const float*)d_in[6];
  const float* attn_r   = (const float*)d_in[7];
  const float* attn_e   = (const float*)d_in[8];
  const float* W_e      = (const float*)d_in[9];
  const float* type_emb = (const float*)d_in[10];
  const float* W_et     = (const float*)d_in[11];
  const float* attn_et  = (const float*)d_in[12];
  const float* bias     = (const float*)d_in[13];
  float* out = (float*)d_out;

  const int N = in_sizes[0] / DIN;   // 50000 (multiple of 16)
  const int E = in_sizes[2];         // 800000

  // Carve workspace (≈80.3 MB total)
  char* p = (char*)d_ws;
  auto carve = [&](size_t bytes) -> char* {
    char* q = p;
    p += (bytes + 255) & ~(size_t)255;
    return q;
  };
  float* fs     = (float*)carve((size_t)N * HF * sizeof(float));
  float* attn   = (float*)carve((size_t)E * H * sizeof(float));
  float* el     = (float*)carve((size_t)N * H * sizeof(float));
  float* er     = (float*)carve((size_t)N * H * sizeof(float));
  int*   rowptr = (int*)  carve((size_t)(N + 1) * sizeof(int));
  float* Me     = (float*)carve(64 * sizeof(float));
  float* ts     = (float*)carve(64 * sizeof(float));
  int*   counts = (int*)  carve(TT * sizeof(int));

  precompute_kernel<<<1, 64, 0, stream>>>(attn_e, W_e, attn_et, W_et,
                                          type_emb, Me, ts, counts);
  hist_kernel<<<(E + 255) / 256, 256, 0, stream>>>(etype, counts, E);
  gemm_wmma_kernel<<<dim3(N / 16, 2), 256, 0, stream>>>(feat, W_src, fs);
  el_er_kernel<<<(N * H + 255) / 256, 256, 0, stream>>>(fs, attn_l, attn_r,
                                                        el, er, N * H);
  edge_attn_kernel<<<(E + 255) / 256, 256, 0, stream>>>(el, er, eemb, src, dst,
                                                        etype, Me, ts, counts,
                                                        attn, E);
  row_ptr_kernel<<<(N + 256) / 256, 256, 0, stream>>>(dst, rowptr, N, E);
  aggregate_kernel<<<N, 256, 0, stream>>>(fs, attn, src, rowptr, bias, out);
}